// CustomAttentionLayer_40742059770518
// MI455X (gfx1250) — compile-verified
//
#include <hip/hip_runtime.h>
#include <math.h>

typedef __attribute__((ext_vector_type(2))) float v2f;
typedef __attribute__((ext_vector_type(8))) float v8f;

#define K_ROWS 2048
#define DIM    512
#define HID4   2048   // 4*DIM
#define EPS_LN 1e-6f
#define RSQRT_D 0.044194173824159216f  // 1/sqrt(512)

// ---------------------------------------------------------------------------
// Prep: g_q = q_iso @ g_w1[:12], g_k = k_iso @ g_w1[12:], sensor distances.
// One wave per row; lane = hidden channel (H=32 == wave32).
// ---------------------------------------------------------------------------
__global__ __launch_bounds__(256) void prep_kernel(
    const float* __restrict__ q_iso, const float* __restrict__ k_iso,
    const float* __restrict__ g_w1,
    const float* __restrict__ qcc, const float* __restrict__ kcc,
    const float* __restrict__ sc,
    float* __restrict__ gq, float* __restrict__ gk,
    float* __restrict__ nks, float* __restrict__ nkps)
{
    const int lane = threadIdx.x & 31;
    const int wave = threadIdx.x >> 5;
    const int row  = blockIdx.x * 8 + wave;     // 0..4095
    const float sx = sc[0], sy = sc[1];
    if (row < K_ROWS) {
        const float* iso = q_iso + row * 12;
        float acc = 0.f;
        #pragma unroll
        for (int c = 0; c < 12; ++c)
            acc = fmaf(iso[c], g_w1[c * 32 + lane], acc);
        gq[row * 32 + lane] = acc;
        if (lane == 0) {
            float dx = qcc[2 * row] - sx, dy = qcc[2 * row + 1] - sy;
            nks[row] = sqrtf(dx * dx + dy * dy);
        }
    } else {
        const int r = row - K_ROWS;
        const float* iso = k_iso + r * 12;
        float acc = 0.f;
        #pragma unroll
        for (int c = 0; c < 12; ++c)
            acc = fmaf(iso[c], g_w1[(12 + c) * 32 + lane], acc);
        gk[r * 32 + lane] = acc;
        if (lane == 0) {
            float dx = kcc[2 * r] - sx, dy = kcc[2 * r + 1] - sy;
            nkps[r] = sqrtf(dx * dx + dy * dy);
        }
    }
}

// ---------------------------------------------------------------------------
// Scores: logits = Q @ K^T / sqrt(D) + R(bias MLPs).
// One wave -> one 16x16 tile via V_WMMA_F32_16X16X4_F32, K-loop over D=512.
// Block = 4 waves = 16(M) x 64(N) strip. grid = (128, 32).
// ---------------------------------------------------------------------------
__global__ __launch_bounds__(128) void scores_kernel(
    const float* __restrict__ query, const float* __restrict__ key,
    const float* __restrict__ qcc, const float* __restrict__ kcc,
    const float* __restrict__ gq, const float* __restrict__ gk,
    const float* __restrict__ nks, const float* __restrict__ nkps,
    const float* __restrict__ g_b1, const float* __restrict__ g_w2,
    const float* __restrict__ g_b2,
    const float* __restrict__ h_w1, const float* __restrict__ h_b1,
    const float* __restrict__ h_w2, const float* __restrict__ h_b2,
    float* __restrict__ logits)
{
    const int lane = threadIdx.x & 31;
    const int wave = threadIdx.x >> 5;
    const int m0 = blockIdx.x * 16;
    const int n0 = (blockIdx.y * 4 + wave) * 16;
    const int nn  = lane & 15;      // M for A-frag, N for B-frag / C-frag
    const int khi = lane >> 4;      // K-half selector

    // --- fp32 WMMA matmul: S = Q K^T over D=512, 4 per step ---
    v8f acc = {};
    const float* arow = query + (m0 + nn) * DIM + 2 * khi;
    const float* brow = key   + (n0 + nn) * DIM + 2 * khi;
    for (int k0 = 0; k0 < DIM; k0 += 4) {
        v2f a = *(const v2f*)(arow + k0);
        v2f b = *(const v2f*)(brow + k0);
        acc = __builtin_amdgcn_wmma_f32_16x16x4_f32(
            false, a, false, b, (short)0, acc, false, false);
    }

    // --- geometric bias R for the 8 (M,N) pairs this lane owns ---
    const int j = n0 + nn;
    const float kcx = kcc[2 * j], kcy = kcc[2 * j + 1];
    const float nkp = nkps[j];
    float nkk[8], nksi[8];
    #pragma unroll
    for (int r = 0; r < 8; ++r) {
        const int i = m0 + 8 * khi + r;
        float dx = qcc[2 * i] - kcx, dy = qcc[2 * i + 1] - kcy;
        nkk[r]  = sqrtf(dx * dx + dy * dy);
        nksi[r] = nks[i];
    }
    float Gacc[8] = {0, 0, 0, 0, 0, 0, 0, 0};
    float Hacc[8] = {0, 0, 0, 0, 0, 0, 0, 0};
    const float* gqbase = gq + (m0 + 8 * khi) * 32;
    const float* gkrow  = gk + j * 32;
    for (int h = 0; h < 32; ++h) {
        const float gb   = g_b1[h];
        const float gw2  = g_w2[h];
        const float hw10 = h_w1[h];
        const float hw11 = h_w1[32 + h];
        const float hw12 = h_w1[64 + h];
        const float hb   = h_b1[h];
        const float hw2v = h_w2[h];
        const float gkv  = gkrow[h];
        const float hpart = fmaf(nkp, hw12, hb);
        #pragma unroll
        for (int r = 0; r < 8; ++r) {
            float gh = gqbase[r * 32 + h] + gkv + gb;
            gh = fmaxf(gh, 0.0f);
            Gacc[r] = fmaf(gh, gw2, Gacc[r]);
            float hh = fmaf(nkk[r], hw10, fmaf(nksi[r], hw11, hpart));
            hh = fmaxf(hh, 0.0f);
            Hacc[r] = fmaf(hh, hw2v, Hacc[r]);
        }
    }
    const float gb2 = g_b2[0], hb2 = h_b2[0];
    #pragma unroll
    for (int r = 0; r < 8; ++r) {
        const float G  = fmaxf(Gacc[r] + gb2, 0.0f);
        const float Hh = Hacc[r] + hb2;
        logits[(m0 + 8 * khi + r) * K_ROWS + j] = acc[r] * RSQRT_D + G * Hh;
    }
}

// ---------------------------------------------------------------------------
// Row softmax in place over 2048 columns. One block (256 threads) per row.
// ---------------------------------------------------------------------------
__global__ __launch_bounds__(256) void softmax_kernel(float* __restrict__ L)
{
    const int tid = threadIdx.x;
    float* p = L + blockIdx.x * K_ROWS;
    __shared__ float red[256];

    float m = -1e30f;
    for (int c = tid; c < K_ROWS; c += 256) m = fmaxf(m, p[c]);
    red[tid] = m; __syncthreads();
    for (int s = 128; s > 0; s >>= 1) {
        if (tid < s) red[tid] = fmaxf(red[tid], red[tid + s]);
        __syncthreads();
    }
    m = red[0]; __syncthreads();

    float sum = 0.f;
    for (int c = tid; c < K_ROWS; c += 256) {
        float e = expf(p[c] - m);
        p[c] = e;
        sum += e;
    }
    red[tid] = sum; __syncthreads();
    for (int s = 128; s > 0; s >>= 1) {
        if (tid < s) red[tid] += red[tid + s];
        __syncthreads();
    }
    const float inv = 1.0f / red[0];
    for (int c = tid; c < K_ROWS; c += 256) p[c] *= inv;
}

// ---------------------------------------------------------------------------
// attn = P @ key  : (2048x2048)@(2048x512). grid = (128, 8), block = 128.
// ---------------------------------------------------------------------------
__global__ __launch_bounds__(128) void pv_kernel(
    const float* __restrict__ P, const float* __restrict__ key,
    float* __restrict__ attn)
{
    const int lane = threadIdx.x & 31;
    const int wave = threadIdx.x >> 5;
    const int m0 = blockIdx.x * 16;
    const int n0 = (blockIdx.y * 4 + wave) * 16;
    const int nn  = lane & 15;
    const int khi = lane >> 4;

    v8f acc = {};
    const float* arow = P + (m0 + nn) * K_ROWS + 2 * khi;
    const float* bcol = key + n0 + nn;
    for (int k0 = 0; k0 < K_ROWS; k0 += 4) {
        v2f a = *(const v2f*)(arow + k0);
        const int kb = k0 + 2 * khi;
        v2f b;
        b.x = bcol[kb * DIM];
        b.y = bcol[(kb + 1) * DIM];
        acc = __builtin_amdgcn_wmma_f32_16x16x4_f32(
            false, a, false, b, (short)0, acc, false, false);
    }
    #pragma unroll
    for (int r = 0; r < 8; ++r)
        attn[(m0 + 8 * khi + r) * DIM + n0 + nn] = acc[r];
}

// ---------------------------------------------------------------------------
// LayerNorm over rows of 512: out = LN(inA (+ inB)) * gamma + beta
// ---------------------------------------------------------------------------
__global__ __launch_bounds__(256) void ln_kernel(
    const float* __restrict__ inA, const float* __restrict__ inB,
    const float* __restrict__ gamma, const float* __restrict__ beta,
    float* __restrict__ out)
{
    const int tid = threadIdx.x;
    const int row = blockIdx.x;
    __shared__ float red[256];

    float v0 = inA[row * DIM + tid];
    float v1 = inA[row * DIM + tid + 256];
    if (inB) {
        v0 += inB[row * DIM + tid];
        v1 += inB[row * DIM + tid + 256];
    }
    red[tid] = v0 + v1; __syncthreads();
    for (int s = 128; s > 0; s >>= 1) {
        if (tid < s) red[tid] += red[tid + s];
        __syncthreads();
    }
    const float mu = red[0] * (1.0f / DIM);
    __syncthreads();
    const float d0 = v0 - mu, d1 = v1 - mu;
    red[tid] = d0 * d0 + d1 * d1; __syncthreads();
    for (int s = 128; s > 0; s >>= 1) {
        if (tid < s) red[tid] += red[tid + s];
        __syncthreads();
    }
    const float rstd = rsqrtf(red[0] * (1.0f / DIM) + EPS_LN);
    out[row * DIM + tid]       = d0 * rstd * gamma[tid]       + beta[tid];
    out[row * DIM + tid + 256] = d1 * rstd * gamma[tid + 256] + beta[tid + 256];
}

// ---------------------------------------------------------------------------
// h1 = relu(x @ W1 + b1) : (2048x512)@(512x2048). grid = (128, 32).
// ---------------------------------------------------------------------------
__global__ __launch_bounds__(128) void ffn1_kernel(
    const float* __restrict__ x, const float* __restrict__ W1,
    const float* __restrict__ b1, float* __restrict__ h1)
{
    const int lane = threadIdx.x & 31;
    const int wave = threadIdx.x >> 5;
    const int m0 = blockIdx.x * 16;
    const int n0 = (blockIdx.y * 4 + wave) * 16;
    const int nn  = lane & 15;
    const int khi = lane >> 4;

    v8f acc = {};
    const float* arow = x + (m0 + nn) * DIM + 2 * khi;
    const float* bcol = W1 + n0 + nn;
    for (int k0 = 0; k0 < DIM; k0 += 4) {
        v2f a = *(const v2f*)(arow + k0);
        const int kb = k0 + 2 * khi;
        v2f b;
        b.x = bcol[kb * HID4];
        b.y = bcol[(kb + 1) * HID4];
        acc = __builtin_amdgcn_wmma_f32_16x16x4_f32(
            false, a, false, b, (short)0, acc, false, false);
    }
    const float bias = b1[n0 + nn];
    #pragma unroll
    for (int r = 0; r < 8; ++r)
        h1[(m0 + 8 * khi + r) * HID4 + n0 + nn] = fmaxf(acc[r] + bias, 0.0f);
}

// ---------------------------------------------------------------------------
// pre = x + h1 @ W2 + b2 : (2048x2048)@(2048x512). grid = (128, 8).
// ---------------------------------------------------------------------------
__global__ __launch_bounds__(128) void ffn2_kernel(
    const float* __restrict__ h1, const float* __restrict__ W2,
    const float* __restrict__ b2, const float* __restrict__ x,
    float* __restrict__ pre)
{
    const int lane = threadIdx.x & 31;
    const int wave = threadIdx.x >> 5;
    const int m0 = blockIdx.x * 16;
    const int n0 = (blockIdx.y * 4 + wave) * 16;
    const int nn  = lane & 15;
    const int khi = lane >> 4;

    v8f acc = {};
    const float* arow = h1 + (m0 + nn) * HID4 + 2 * khi;
    const float* bcol = W2 + n0 + nn;
    for (int k0 = 0; k0 < HID4; k0 += 4) {
        v2f a = *(const v2f*)(arow + k0);
        const int kb = k0 + 2 * khi;
        v2f b;
        b.x = bcol[kb * DIM];
        b.y = bcol[(kb + 1) * DIM];
        acc = __builtin_amdgcn_wmma_f32_16x16x4_f32(
            false, a, false, b, (short)0, acc, false, false);
    }
    const float bias = b2[n0 + nn];
    #pragma unroll
    for (int r = 0; r < 8; ++r) {
        const int idx = (m0 + 8 * khi + r) * DIM + n0 + nn;
        pre[idx] = x[idx] + acc[r] + bias;
    }
}

// ---------------------------------------------------------------------------
extern "C" void kernel_launch(void* const* d_in, const int* in_sizes, int n_in,
                              void* d_out, int out_size, void* d_ws, size_t ws_size,
                              hipStream_t stream)
{
    const float* sc    = (const float*)d_in[0];
    const float* query = (const float*)d_in[1];
    const float* key   = (const float*)d_in[2];
    const float* qcc   = (const float*)d_in[3];
    const float* kcc   = (const float*)d_in[4];
    const float* q_iso = (const float*)d_in[5];
    const float* k_iso = (const float*)d_in[6];
    const float* g_w1  = (const float*)d_in[7];
    const float* g_b1  = (const float*)d_in[8];
    const float* g_w2  = (const float*)d_in[9];
    const float* g_b2  = (const float*)d_in[10];
    const float* h_w1  = (const float*)d_in[11];
    const float* h_b1  = (const float*)d_in[12];
    const float* h_w2  = (const float*)d_in[13];
    const float* h_b2  = (const float*)d_in[14];
    const float* w1    = (const float*)d_in[15];
    const float* b1    = (const float*)d_in[16];
    const float* w2    = (const float*)d_in[17];
    const float* b2    = (const float*)d_in[18];
    const float* n1g   = (const float*)d_in[19];
    const float* n1b   = (const float*)d_in[20];
    const float* n2g   = (const float*)d_in[21];
    const float* n2b   = (const float*)d_in[22];
    float* out = (float*)d_out;

    // workspace carve-up (floats); everything fits in ~30 MB -> L2 resident
    float* ws     = (float*)d_ws;
    float* gq     = ws;                          // 2048*32
    float* gk     = gq + K_ROWS * 32;            // 2048*32
    float* nks    = gk + K_ROWS * 32;            // 2048
    float* nkps   = nks + K_ROWS;                // 2048
    float* logits = nkps + K_ROWS;               // 2048*2048 (reused as h1)
    float* attn   = logits + K_ROWS * K_ROWS;    // 2048*512
    float* x      = attn + K_ROWS * DIM;         // 2048*512
    float* pre    = x + K_ROWS * DIM;            // 2048*512

    prep_kernel<<<512, 256, 0, stream>>>(q_iso, k_iso, g_w1, qcc, kcc, sc,
                                         gq, gk, nks, nkps);
    scores_kernel<<<dim3(128, 32), 128, 0, stream>>>(
        query, key, qcc, kcc, gq, gk, nks, nkps,
        g_b1, g_w2, g_b2, h_w1, h_b1, h_w2, h_b2, logits);
    softmax_kernel<<<2048, 256, 0, stream>>>(logits);
    pv_kernel<<<dim3(128, 8), 128, 0, stream>>>(logits, key, attn);
    ln_kernel<<<2048, 256, 0, stream>>>(query, attn, n1g, n1b, x);
    ffn1_kernel<<<dim3(128, 32), 128, 0, stream>>>(x, w1, b1, logits /*h1*/);
    ffn2_kernel<<<dim3(128, 8), 128, 0, stream>>>(logits, w2, b2, x, pre);
    ln_kernel<<<2048, 256, 0, stream>>>(pre, nullptr, n2g, n2b, out);
}